// SingleStageFIRFilter_29824252903954
// MI455X (gfx1250) — compile-verified
//
#include <hip/hip_runtime.h>
#include <math.h>

typedef float v2f __attribute__((ext_vector_type(2)));
typedef float v8f __attribute__((ext_vector_type(8)));

#define BATCH   2
#define T_LEN   16000
#define NCEP    25      // mc has M+1 = 25 coefficients
#define M_ORD   24
#define IR_LEN  2000
#define TILE_T  128     // time steps per workgroup
#define NWAVES  4       // 128 threads = 4 wave32
#define CHUNK   80      // taps per chunk (2000 = 25 * 80)
#define NCHUNK  25
#define HSTRIDE 81      // 80 + 1 pad -> gcd(81,64)=1, conflict-free LDS
#define XS_LEN  2176    // >= (IR_LEN-1) + TILE_T = 2127, padded

// y[b,t] = sum_{k=0..1999} h[b,t,k] * x[b,t-k]
// h[0] = exp(c0); h[n] = (1/n) * sum_{k=1..24} k*c_k*h[n-k]   (c2ir recursion)
__global__ __launch_bounds__(TILE_T)
void tv_fir_c2ir_kernel(const float* __restrict__ x,
                        const float* __restrict__ mc,
                        float* __restrict__ y) {
  __shared__ float xs[XS_LEN];                 // x window (zero padded history)
  __shared__ float hs[TILE_T * HSTRIDE];       // h chunk: [t_local][tap_local]
  __shared__ float invn[IR_LEN];               // 1/n table
  __shared__ float dbuf[NWAVES][16][16];       // WMMA D-tile staging per wave

  const int tid  = threadIdx.x;
  const int wave = tid >> 5;
  const int lane = tid & 31;
  const int hi   = lane >> 4;                  // 0: lanes 0-15, 1: lanes 16-31
  const int lm   = lane & 15;

  const int b  = blockIdx.x / (T_LEN / TILE_T);
  const int tb = (blockIdx.x % (T_LEN / TILE_T)) * TILE_T;

  // --- load x window: xs[i] = x[b, tb - 1999 + i] (0 outside batch) ---
  for (int i = tid; i < XS_LEN; i += TILE_T) {
    int g = tb - (IR_LEN - 1) + i;
    xs[i] = (i < (IR_LEN - 1 + TILE_T) && g >= 0 && g < T_LEN)
                ? x[(size_t)b * T_LEN + g] : 0.0f;
  }
  // --- 1/n table (n uniform per tap; one LDS broadcast in the hot loop) ---
  for (int n = tid; n < IR_LEN; n += TILE_T)
    invn[n] = (n > 0) ? 1.0f / (float)n : 0.0f;

  // --- per-lane cepstrum for chain t = tb + tid ---
  const float* mcp = mc + ((size_t)b * T_LEN + (tb + tid)) * NCEP;
  float c0 = mcp[0];
  float kc[NCEP];                              // kc[k] = k * c_k
#pragma unroll
  for (int k = 1; k < NCEP; ++k) kc[k] = (float)k * mcp[k];

  v8f acc0 = {};                               // 16x16 f32 WMMA accumulators
  v8f acc1 = {};
  float hp[M_ORD];                             // rolling window h[n-1..n-24]
#pragma unroll
  for (int k = 0; k < M_ORD; ++k) hp[k] = 0.0f;

  __syncthreads();

  float* hrow = &hs[tid * HSTRIDE];

  for (int chunk = 0; chunk < NCHUNK; ++chunk) {
    const int n0 = chunk * CHUNK;

    // ---- recursion phase: each lane produces CHUNK taps of its chain ----
    if (chunk == 0) {
      float h0 = __expf(c0);
      hrow[0] = h0;
#pragma unroll
      for (int k2 = M_ORD - 1; k2 > 0; --k2) hp[k2] = hp[k2 - 1];
      hp[0] = h0;
#pragma unroll 10
      for (int j = 1; j < CHUNK; ++j) {
        float s = 0.0f;
#pragma unroll
        for (int k = 1; k <= M_ORD; ++k) s += kc[k] * hp[k - 1];
        float hn = s * invn[j];
        hrow[j] = hn;
#pragma unroll
        for (int k2 = M_ORD - 1; k2 > 0; --k2) hp[k2] = hp[k2 - 1];
        hp[0] = hn;
      }
    } else {
#pragma unroll 10
      for (int j = 0; j < CHUNK; ++j) {
        float s = 0.0f;
#pragma unroll
        for (int k = 1; k <= M_ORD; ++k) s += kc[k] * hp[k - 1];
        float hn = s * invn[n0 + j];
        hrow[j] = hn;
#pragma unroll
        for (int k2 = M_ORD - 1; k2 > 0; --k2) hp[k2] = hp[k2 - 1];
        hp[0] = hn;
      }
    }
    __syncthreads();

    // ---- WMMA FIR phase: D[i][n] += sum_j h[t0+i,k0+j] * x[t0+n-k0-j] ----
    // A (16x4 f32): lane -> M=lane%16, vector elem r -> K = r + 2*(lane/16)
    // B (4x16 f32): lane -> N=lane%16, vector elem r -> row j = r + 2*(lane/16)
#pragma unroll
    for (int st = 0; st < 2; ++st) {
      const int sub = wave * 32 + st * 16;     // subtile t-offset in tile
      v8f acc = st ? acc1 : acc0;
      const float* hsub = &hs[(sub + lm) * HSTRIDE];
#pragma unroll
      for (int g = 0; g < CHUNK / 4; ++g) {
        const int kl = g * 4 + 2 * hi;         // local tap of A.x / B row 0
        v2f a;  a.x = hsub[kl];  a.y = hsub[kl + 1];
        const int xb = (IR_LEN - 1) + sub + lm - (n0 + kl);
        v2f bv; bv.x = xs[xb];   bv.y = xs[xb - 1];
        acc = __builtin_amdgcn_wmma_f32_16x16x4_f32(
            /*neg_a=*/false, a, /*neg_b=*/false, bv,
            /*c_mod=*/(short)0, acc, /*reuse_a=*/false, /*reuse_b=*/false);
      }
      if (st) acc1 = acc; else acc0 = acc;
    }
    __syncthreads();
  }

  // ---- diagonal extraction: y[t0+i] = D[i][i] ----
  for (int st = 0; st < 2; ++st) {
    v8f acc = st ? acc1 : acc0;
#pragma unroll
    for (int r = 0; r < 8; ++r)
      dbuf[wave][r + 8 * hi][lm] = acc[r];     // C/D layout: M=r+8*hi, N=lane%16
    __syncthreads();
    if (lane < 16) {
      const int sub = wave * 32 + st * 16;
      y[(size_t)b * T_LEN + tb + sub + lane] = dbuf[wave][lane][lane];
    }
    __syncthreads();
  }
}

extern "C" void kernel_launch(void* const* d_in, const int* in_sizes, int n_in,
                              void* d_out, int out_size, void* d_ws, size_t ws_size,
                              hipStream_t stream) {
  (void)in_sizes; (void)n_in; (void)out_size; (void)d_ws; (void)ws_size;
  const float* x  = (const float*)d_in[0];   // (2, 16000) f32
  const float* mc = (const float*)d_in[1];   // (2, 16000, 25) f32
  float* y = (float*)d_out;                  // (2, 16000) f32

  dim3 grid(BATCH * (T_LEN / TILE_T));       // 250 workgroups
  dim3 block(TILE_T);                        // 128 threads = 4 wave32
  tv_fir_c2ir_kernel<<<grid, block, 0, stream>>>(x, mc, y);
}